// dqgnLight_20057497272726
// MI455X (gfx1250) — compile-verified
//
#include <hip/hip_runtime.h>

// CDNA5 / gfx1250: wave32, WMMA 16x16x4 f32 (fp32 precision matches reference).
// Roofline: dominant cost is streaming Wq (33.5 MB) once ~= 1.4us @ 23.3 TB/s.
// All GEMM work is expressed as V_WMMA_F32_16X16X4_F32.

typedef __attribute__((ext_vector_type(2))) float v2f;
typedef __attribute__((ext_vector_type(8))) float v8f;

constexpr int N = 256;   // nodes
constexpr int F = 16;    // features
constexpr int H = 2048;  // hidden
constexpr int P = 16;    // phases

// ---------------- graph preprocessing (tiny) ----------------

__global__ void k_zero(float* p, int n) {
  int i = blockIdx.x * blockDim.x + threadIdx.x;
  if (i < n) p[i] = 0.0f;
}

// deg[dst] += 1.0 per edge (exact integer sums -> deterministic)
__global__ void k_degree(const int* __restrict__ ei, int E, float* __restrict__ deg) {
  int e = blockIdx.x * blockDim.x + threadIdx.x;
  if (e < E) atomicAdd(&deg[ei[E + e]], 1.0f);
}

// dinv = rsqrt(deg + 1) (self loop); diagonal of A = dinv^2 (self-loop msg)
__global__ void k_dinv_diag(const float* __restrict__ deg, float* __restrict__ dinv,
                            float* __restrict__ A) {
  int n = threadIdx.x;
  float di = rsqrtf(deg[n] + 1.0f);
  dinv[n] = di;
  A[n * N + n] = di * di;
}

// A[dst,src] += dinv[src]*dinv[dst]
__global__ void k_adj(const int* __restrict__ ei, int E, const float* __restrict__ dinv,
                      float* __restrict__ A) {
  int e = blockIdx.x * blockDim.x + threadIdx.x;
  if (e < E) {
    int s = ei[e], d = ei[E + e];
    atomicAdd(&A[d * N + s], dinv[s] * dinv[d]);
  }
}

// ---------------- Ax = A @ x  ([256,256] x [256,16]) ----------------
// One wave per 16-row M tile; K=256 -> 64 WMMA_F32_16X16X4_F32 steps.
__global__ void k_ax(const float* __restrict__ A, const float* __restrict__ x,
                     float* __restrict__ Ax) {
  int wave = blockIdx.x * (blockDim.x >> 5) + (threadIdx.x >> 5);
  int lane = threadIdx.x & 31;
  int half = lane >> 4, l = lane & 15;
  int m0 = wave * 16;
  v8f c0 = {}, c1 = {};
  for (int k0 = 0; k0 < N; k0 += 8) {
    v2f a, bm;
    // A-operand: lanes 0-15 K={k0,k0+1}, lanes 16-31 K={k0+2,k0+3}
    const float* ap = &A[(m0 + l) * N + k0 + 2 * half];
    a.x = ap[0]; a.y = ap[1];
    bm.x = x[(k0 + 2 * half) * F + l];
    bm.y = x[(k0 + 2 * half + 1) * F + l];
    c0 = __builtin_amdgcn_wmma_f32_16x16x4_f32(false, a, false, bm, (short)0, c0, false, false);
    const float* ap2 = &A[(m0 + l) * N + k0 + 4 + 2 * half];
    a.x = ap2[0]; a.y = ap2[1];
    bm.x = x[(k0 + 4 + 2 * half) * F + l];
    bm.y = x[(k0 + 5 + 2 * half) * F + l];
    c1 = __builtin_amdgcn_wmma_f32_16x16x4_f32(false, a, false, bm, (short)0, c1, false, false);
  }
#pragma unroll
  for (int v = 0; v < 8; ++v)
    Ax[(m0 + v + 8 * half) * F + l] = c0[v] + c1[v];
}

// ---------------- out = Ax @ W + b  ([256,16] x [16,2048]) ----------------
// 16 x 128 tiles of 16x16; one wave per tile; K=16 -> 4 WMMA steps; bias in C.
__global__ void k_hidden(const float* __restrict__ Ax, const float* __restrict__ W,
                         const float* __restrict__ bias, float* __restrict__ outv) {
  int wave = blockIdx.x * (blockDim.x >> 5) + (threadIdx.x >> 5);  // 0..2047
  int lane = threadIdx.x & 31;
  int half = lane >> 4, l = lane & 15;
  int mt = wave >> 7, nt = wave & 127;
  int m0 = mt * 16, n0 = nt * 16;
  float cb = bias[n0 + l];
  v8f c;
#pragma unroll
  for (int v = 0; v < 8; ++v) c[v] = cb;
#pragma unroll
  for (int k0 = 0; k0 < F; k0 += 4) {
    v2f a, bm;
    const float* ap = &Ax[(m0 + l) * F + k0 + 2 * half];
    a.x = ap[0]; a.y = ap[1];
    bm.x = W[(k0 + 2 * half) * H + n0 + l];
    bm.y = W[(k0 + 2 * half + 1) * H + n0 + l];
    c = __builtin_amdgcn_wmma_f32_16x16x4_f32(false, a, false, bm, (short)0, c, false, false);
  }
#pragma unroll
  for (int v = 0; v < 8; ++v)
    outv[(m0 + v + 8 * half) * H + n0 + l] = c[v];
}

// ---------------- q[n] = out[n] . Wq[n] + bq[n]  (dominant: streams Wq once) ----------------
// Block = node (256 blocks x 8 waves). Wave w covers h in [w*256, w*256+256).
// WMMA: A = Wq[n]^T chunk (M=p=lane%16, K=4 h values -> coalesced 64B Wq rows),
//       B = out[n,h] broadcast across all 16 columns -> D columns identical;
//       lane0/lane16 of each acc VGPR carry q[p]. Deterministic LDS reduction.
__global__ void k_q(const float* __restrict__ outv, const float* __restrict__ Wq,
                    const float* __restrict__ bq, float* __restrict__ q) {
  int n = blockIdx.x;
  int w = threadIdx.x >> 5;
  int lane = threadIdx.x & 31;
  int half = lane >> 4, l = lane & 15;
  const float* wqn = Wq + (size_t)n * H * P;
  const float* on  = outv + n * H;
  int hbase = w * (H / 8);  // 256 h-values per wave
  v8f c0 = {}, c1 = {};
  for (int s = 0; s < (H / 8); s += 8) {  // 32 iterations, 2 WMMAs each
    int h0 = hbase + s;
    v2f a, bm;
    a.x = wqn[(h0 + 2 * half) * P + l];
    a.y = wqn[(h0 + 2 * half + 1) * P + l];
    bm.x = on[h0 + 2 * half];
    bm.y = on[h0 + 2 * half + 1];
    c0 = __builtin_amdgcn_wmma_f32_16x16x4_f32(false, a, false, bm, (short)0, c0, false, false);
    a.x = wqn[(h0 + 4 + 2 * half) * P + l];
    a.y = wqn[(h0 + 5 + 2 * half) * P + l];
    bm.x = on[h0 + 4 + 2 * half];
    bm.y = on[h0 + 5 + 2 * half];
    c1 = __builtin_amdgcn_wmma_f32_16x16x4_f32(false, a, false, bm, (short)0, c1, false, false);
  }
  __shared__ float qp[8][16];
  if (lane == 0) {
#pragma unroll
    for (int v = 0; v < 8; ++v) qp[w][v] = c0[v] + c1[v];        // p = v
  } else if (lane == 16) {
#pragma unroll
    for (int v = 0; v < 8; ++v) qp[w][8 + v] = c0[v] + c1[v];    // p = v + 8
  }
  __syncthreads();
  if (threadIdx.x < 16) {
    float s = bq[n * P + threadIdx.x];
#pragma unroll
    for (int ww = 0; ww < 8; ++ww) s += qp[ww][threadIdx.x];
    q[n * P + threadIdx.x] = s;
  }
}

extern "C" void kernel_launch(void* const* d_in, const int* in_sizes, int n_in,
                              void* d_out, int out_size, void* d_ws, size_t ws_size,
                              hipStream_t stream) {
  const float* x    = (const float*)d_in[0];
  const int*   ei   = (const int*)d_in[1];
  const float* W    = (const float*)d_in[2];
  const float* bias = (const float*)d_in[3];
  const float* Wq   = (const float*)d_in[4];
  const float* bq   = (const float*)d_in[5];
  float* q = (float*)d_out;
  int E = in_sizes[1] / 2;

  // workspace layout (floats): A[N*N] | deg[N] | dinv[N] | Ax[N*F] | out[N*H]
  float* wsf  = (float*)d_ws;
  float* A    = wsf;
  float* deg  = A + N * N;
  float* dinv = deg + N;
  float* Ax   = dinv + N;
  float* outv = Ax + N * F;

  int zn = N * N + N;  // zero A and deg (contiguous); Ax/outv fully overwritten
  k_zero<<<(zn + 255) / 256, 256, 0, stream>>>(wsf, zn);
  k_degree<<<(E + 255) / 256, 256, 0, stream>>>(ei, E, deg);
  k_dinv_diag<<<1, N, 0, stream>>>(deg, dinv, A);
  k_adj<<<(E + 255) / 256, 256, 0, stream>>>(ei, E, dinv, A);
  k_ax<<<2, 256, 0, stream>>>(A, x, Ax);            // 16 waves = 16 M-tiles
  k_hidden<<<256, 256, 0, stream>>>(Ax, W, bias, outv);  // 2048 waves = 2048 tiles
  k_q<<<N, 256, 0, stream>>>(outv, Wq, bq, q);      // 1 block per node, 8 waves
}